// Net_71330816852788
// MI455X (gfx1250) — compile-verified
//
#include <hip/hip_runtime.h>
#include <hip/hip_bf16.h>
#include <math.h>

// ---------------------------------------------------------------------------
// GCS-GNN pipeline for MI455X (gfx1250, wave32).
//   3x GCSConv(CH=32) -> segment mean pool -> sigmoid(dense)
// GEMMs use V_WMMA_F32_16X16X4_F32 (f32 tensor path, ISA 7.12.2 layouts).
// Edge scatter uses relaxed agent-scope f32 atomics (L2-resident acc).
// ---------------------------------------------------------------------------

#define NNODES 262144
#define NEDGES 4194304
#define NGRAPH 4096
#define CH     32

typedef float v2f __attribute__((ext_vector_type(2)));
typedef float v8f __attribute__((ext_vector_type(8)));

// ---------------- layer 1: DIN=1 outer product ------------------------------
// h[i][c] = x[i]*W1[c];  acc[i][c] = x[i]*W2[c] + b[c]
__global__ void layer1_kernel(const float* __restrict__ x,
                              const float* __restrict__ W1,
                              const float* __restrict__ W2,
                              const float* __restrict__ b,
                              float* __restrict__ h,
                              float* __restrict__ acc)
{
    int gid = blockIdx.x * blockDim.x + threadIdx.x;   // N*32 threads
    if (gid >= NNODES * CH) return;
    int i = gid >> 5;
    int c = gid & 31;
    float xv = x[i];                 // 32 consecutive lanes share x[i]
    h[gid]   = xv * W1[c];
    acc[gid] = fmaf(xv, W2[c], b[c]);
}

// ---------------- WMMA dual-GEMM: h = f(x)@W1, acc = f(x)@W2 + b ------------
// f = relu if relu_in. One wave per 16-row tile. K=32 in 8 steps of K=4.
__global__ void gcs_gemm_kernel(const float* __restrict__ xin,   // [N,32]
                                const float* __restrict__ W1,    // [32,32]
                                const float* __restrict__ W2,    // [32,32]
                                const float* __restrict__ bias,  // [32]
                                float* __restrict__ h,           // [N,32]
                                float* __restrict__ acc,         // [N,32]
                                int relu_in)
{
    const int lane = threadIdx.x & 31;
    const int wave = threadIdx.x >> 5;
    const int tile = blockIdx.x * 8 + wave;          // 16-row tile (wave-uniform)
    const int row0 = tile * 16;
    if (row0 >= NNODES) return;                      // uniform per wave: EXEC stays all-1s

    const int l16  = lane & 15;
    const int hi   = lane >> 4;                      // 0 | 1
    const int koff = hi * 2;                         // f32 A/B lane split per ISA 7.12.2

    // B fragments (4x16 f32, VGPR0->k0, VGPR1->k0+1; k0 = 2*hi) for both weights.
    v2f bf1[2][8], bf2[2][8];
#pragma unroll
    for (int ct = 0; ct < 2; ++ct) {
        const int ncol = ct * 16 + l16;
#pragma unroll
        for (int kk = 0; kk < 8; ++kk) {
            const int k0 = kk * 4 + koff;
            bf1[ct][kk].x = W1[k0 * CH + ncol];
            bf1[ct][kk].y = W1[(k0 + 1) * CH + ncol];
            bf2[ct][kk].x = W2[k0 * CH + ncol];
            bf2[ct][kk].y = W2[(k0 + 1) * CH + ncol];
        }
    }

    // Accumulators: c1 starts at 0, c2 starts at bias[n] (bias along N dim).
    v8f c1[2], c2[2];
#pragma unroll
    for (int ct = 0; ct < 2; ++ct) {
        const float bv = bias[ct * 16 + l16];
#pragma unroll
        for (int r = 0; r < 8; ++r) { c1[ct][r] = 0.0f; c2[ct][r] = bv; }
    }

    // A fragments: lane handles row row0+l16; lane<16 -> K{0,1}, lane>=16 -> K{2,3}.
    const float* xrow = xin + (size_t)(row0 + l16) * CH;
    v2f a[8];
#pragma unroll
    for (int kk = 0; kk < 8; ++kk) {
        float ax = xrow[kk * 4 + koff];
        float ay = xrow[kk * 4 + koff + 1];
        if (relu_in) { ax = fmaxf(ax, 0.0f); ay = fmaxf(ay, 0.0f); }
        a[kk].x = ax; a[kk].y = ay;
    }

#pragma unroll
    for (int kk = 0; kk < 8; ++kk) {
        c1[0] = __builtin_amdgcn_wmma_f32_16x16x4_f32(false, a[kk], false, bf1[0][kk],
                                                      (short)0, c1[0], false, false);
        c1[1] = __builtin_amdgcn_wmma_f32_16x16x4_f32(false, a[kk], false, bf1[1][kk],
                                                      (short)0, c1[1], false, false);
        c2[0] = __builtin_amdgcn_wmma_f32_16x16x4_f32(false, a[kk], false, bf2[0][kk],
                                                      (short)0, c2[0], false, false);
        c2[1] = __builtin_amdgcn_wmma_f32_16x16x4_f32(false, a[kk], false, bf2[1][kk],
                                                      (short)0, c2[1], false, false);
    }

    // Store: C element (M = r + 8*hi, N = ct*16 + l16).
#pragma unroll
    for (int ct = 0; ct < 2; ++ct) {
#pragma unroll
        for (int r = 0; r < 8; ++r) {
            const int m = r + 8 * hi;
            const size_t idx = (size_t)(row0 + m) * CH + ct * 16 + l16;
            h[idx]   = c1[ct][r];
            acc[idx] = c2[ct][r];
        }
    }
}

// ---------------- edge scatter: acc[dst] += w * h[src] ----------------------
// 8 lanes per edge, float4 gather (L2-resident h), 4 f32 atomics each.
__global__ void edge_kernel(const int*   __restrict__ src,
                            const int*   __restrict__ dst,
                            const float* __restrict__ w,
                            const float* __restrict__ h,
                            float*       __restrict__ acc)
{
    long long gid = (long long)blockIdx.x * blockDim.x + threadIdx.x;  // E*8 threads
    if (gid >= (long long)NEDGES * 8) return;
    const int e = (int)(gid >> 3);
    const int g = ((int)gid & 7) * 4;
    const int s = src[e];
    const int d = dst[e];
    const float wt = w[e];
    const float4 hv = *(const float4*)(h + (size_t)s * CH + g);
    float* ap = acc + (size_t)d * CH + g;
    __hip_atomic_fetch_add(ap + 0, wt * hv.x, __ATOMIC_RELAXED, __HIP_MEMORY_SCOPE_AGENT);
    __hip_atomic_fetch_add(ap + 1, wt * hv.y, __ATOMIC_RELAXED, __HIP_MEMORY_SCOPE_AGENT);
    __hip_atomic_fetch_add(ap + 2, wt * hv.z, __ATOMIC_RELAXED, __HIP_MEMORY_SCOPE_AGENT);
    __hip_atomic_fetch_add(ap + 3, wt * hv.w, __ATOMIC_RELAXED, __HIP_MEMORY_SCOPE_AGENT);
}

// ---------------- zero sums/counts (replay-safe) ----------------------------
__global__ void zero_kernel(float* __restrict__ sums, float* __restrict__ counts)
{
    int gid = blockIdx.x * blockDim.x + threadIdx.x;
    if (gid < NGRAPH * CH) sums[gid] = 0.0f;
    if (gid < NGRAPH)      counts[gid] = 0.0f;
}

// ---------------- segment pooling: sums[seg[i]] += relu(acc[i]) -------------
__global__ void pool_kernel(const float* __restrict__ acc,
                            const int*   __restrict__ seg,
                            float* __restrict__ sums,
                            float* __restrict__ counts)
{
    int gid = blockIdx.x * blockDim.x + threadIdx.x;   // N*32 threads
    if (gid >= NNODES * CH) return;
    const int i = gid >> 5;
    const int c = gid & 31;
    const int g = seg[i];
    const float v = fmaxf(acc[gid], 0.0f);
    __hip_atomic_fetch_add(&sums[g * CH + c], v, __ATOMIC_RELAXED, __HIP_MEMORY_SCOPE_AGENT);
    if (c == 0)
        __hip_atomic_fetch_add(&counts[g], 1.0f, __ATOMIC_RELAXED, __HIP_MEMORY_SCOPE_AGENT);
}

// ---------------- head: sigmoid(mean_pool @ Wd + bd) ------------------------
__global__ void head_kernel(const float* __restrict__ sums,
                            const float* __restrict__ counts,
                            const float* __restrict__ Wd,
                            const float* __restrict__ bd,
                            float* __restrict__ out)
{
    int g = blockIdx.x * blockDim.x + threadIdx.x;
    if (g >= NGRAPH) return;
    const float inv = 1.0f / fmaxf(counts[g], 1.0f);
    float acc = bd[0];
#pragma unroll
    for (int c = 0; c < CH; ++c)
        acc = fmaf(sums[g * CH + c] * inv, Wd[c], acc);
    out[g] = 1.0f / (1.0f + expf(-acc));
}

// ---------------------------------------------------------------------------
extern "C" void kernel_launch(void* const* d_in, const int* in_sizes, int n_in,
                              void* d_out, int out_size, void* d_ws, size_t ws_size,
                              hipStream_t stream)
{
    const float* x        = (const float*)d_in[0];
    const int*   edge_src = (const int*)  d_in[1];
    const int*   edge_dst = (const int*)  d_in[2];
    const float* edge_w   = (const float*)d_in[3];
    const int*   seg      = (const int*)  d_in[4];
    const float* W1_1 = (const float*)d_in[5];
    const float* W2_1 = (const float*)d_in[6];
    const float* b1   = (const float*)d_in[7];
    const float* W1_2 = (const float*)d_in[8];
    const float* W2_2 = (const float*)d_in[9];
    const float* b2   = (const float*)d_in[10];
    const float* W1_3 = (const float*)d_in[11];
    const float* W2_3 = (const float*)d_in[12];
    const float* b3   = (const float*)d_in[13];
    const float* Wd   = (const float*)d_in[14];
    const float* bd   = (const float*)d_in[15];
    float* out = (float*)d_out;

    // Workspace layout (floats): h | accA | accB | sums | counts
    float* ws     = (float*)d_ws;
    const size_t NC = (size_t)NNODES * CH;
    float* h      = ws;
    float* accA   = ws + NC;
    float* accB   = ws + 2 * NC;
    float* sums   = ws + 3 * NC;
    float* counts = ws + 3 * NC + (size_t)NGRAPH * CH;

    const int BT = 256;
    const int gridNC   = (NNODES * CH + BT - 1) / BT;            // node x channel
    const int gridEdge = (int)(((long long)NEDGES * 8 + BT - 1) / BT);
    const int gridGemm = NNODES / 16 / 8;                        // 8 waves/block
    const int gridZero = (NGRAPH * CH + BT - 1) / BT;
    const int gridHead = (NGRAPH + BT - 1) / BT;

    zero_kernel<<<gridZero, BT, 0, stream>>>(sums, counts);

    // ---- layer 1 (DIN=1) ----
    layer1_kernel<<<gridNC, BT, 0, stream>>>(x, W1_1, W2_1, b1, h, accA);
    edge_kernel<<<gridEdge, BT, 0, stream>>>(edge_src, edge_dst, edge_w, h, accA);

    // ---- layer 2 (relu fused into A-load) ----
    gcs_gemm_kernel<<<gridGemm, BT, 0, stream>>>(accA, W1_2, W2_2, b2, h, accB, 1);
    edge_kernel<<<gridEdge, BT, 0, stream>>>(edge_src, edge_dst, edge_w, h, accB);

    // ---- layer 3 ----
    gcs_gemm_kernel<<<gridGemm, BT, 0, stream>>>(accB, W1_3, W2_3, b3, h, accA, 1);
    edge_kernel<<<gridEdge, BT, 0, stream>>>(edge_src, edge_dst, edge_w, h, accA);

    // ---- pool + head (relu fused into pool load) ----
    pool_kernel<<<gridNC, BT, 0, stream>>>(accA, seg, sums, counts);
    head_kernel<<<gridHead, BT, 0, stream>>>(sums, counts, Wd, bd, out);
}